// BipartiteKernel_30897994727580
// MI455X (gfx1250) — compile-verified
//
#include <hip/hip_runtime.h>
#include <hip/hip_bf16.h>
#include <math.h>

#define N_OBS   100000
#define N_QRY   50000
#define N_EDGE  800000
#define LATENT  128
#define HEADS   4
#define HEAD_DIM 32

typedef float v2f __attribute__((ext_vector_type(2)));
typedef float v4f __attribute__((ext_vector_type(4)));
typedef float v8f __attribute__((ext_vector_type(8)));

// ---------------------------------------------------------------------------
// Kernel 1: v = h_obs @ Wv + bv   (f32 WMMA 16x16x4)
// One 16x16 output tile per wave; 8 waves -> 16x128 row block per workgroup.
// A layout (16x4 f32): lane L holds A[m=L&15][k = 2*(L>>4) + {0,1}]
// B layout (4x16 f32): lane L holds B[k = 2*(L>>4) + {0,1}][n=L&15]
// C/D layout: VGPR r -> row (r + 8*(L>>4)), col (L&15)
// ---------------------------------------------------------------------------
__global__ __launch_bounds__(256) void value_proj_kernel(
    const float* __restrict__ h_obs, const float* __restrict__ Wv,
    const float* __restrict__ bv, float* __restrict__ v)
{
  const int wave = threadIdx.x >> 5;
  const int lane = threadIdx.x & 31;
  const int ph   = lane >> 4;          // 0 or 1 (which lane half)
  const int mn   = lane & 15;
  const int r0   = ph * 2;             // k sub-pair base
  const int rowBase = blockIdx.x * 16;
  const int colBase = wave * 16;

  v8f acc = {};
  const float* arow = h_obs + (size_t)(rowBase + mn) * LATENT;
  #pragma unroll 4
  for (int k0 = 0; k0 < LATENT; k0 += 4) {
    v2f a = *(const v2f*)(arow + k0 + r0);
    v2f b;
    b.x = Wv[(size_t)(k0 + r0) * LATENT + colBase + mn];
    b.y = Wv[(size_t)(k0 + r0 + 1) * LATENT + colBase + mn];
    acc = __builtin_amdgcn_wmma_f32_16x16x4_f32(false, a, false, b,
                                                (short)0, acc, false, false);
  }
  const float bias = bv[colBase + mn];
  #pragma unroll
  for (int r = 0; r < 8; ++r) {
    int row = rowBase + r + 8 * ph;
    v[(size_t)row * LATENT + colBase + mn] = acc[r] + bias;
  }
}

// ---------------------------------------------------------------------------
// Kernel 2: per-edge MLP, both layers on WMMA.
// 16 edges per wave, 8 waves per block, E/128 = 6250 blocks.
// Layer 1: [rel,1](16x4) @ [W1;b1](4x128) -> relu -> staged in LDS.
// Layer 2: hid(16x128) @ W2pad(128x16, cols>=4 zero) -> logits (16x4).
// Also builds dst-histogram with int atomics.
// ---------------------------------------------------------------------------
__global__ __launch_bounds__(256) void edge_mlp_kernel(
    const float* __restrict__ pos_obs, const float* __restrict__ pos_query,
    const int* __restrict__ src, const int* __restrict__ dst,
    const float* __restrict__ W1, const float* __restrict__ b1,
    const float* __restrict__ W2, const float* __restrict__ b2,
    float* __restrict__ logits, int* __restrict__ cnt)
{
  __shared__ float hidbuf[8 * 16 * 136];   // 16 rows x 136-float padded stride per wave
  const int wave = threadIdx.x >> 5;
  const int lane = threadIdx.x & 31;
  const int ph   = lane >> 4;
  const int mn   = lane & 15;
  const int r0   = ph * 2;
  const int e0   = (blockIdx.x * 8 + wave) * 16;
  float* myhid = hidbuf + wave * (16 * 136);

  // rel for edge e0+mn (both lane halves load the same edge's data)
  const int e = e0 + mn;
  const int s = src[e];
  const int d = dst[e];
  const float rx = pos_query[d * 3 + 0] - pos_obs[s * 3 + 0];
  const float ry = pos_query[d * 3 + 1] - pos_obs[s * 3 + 1];
  const float rz = pos_query[d * 3 + 2] - pos_obs[s * 3 + 2];
  v2f a1;
  a1.x = ph ? rz : rx;     // k rows: 0=x, 1=y, 2=z, 3=1 (bias row)
  a1.y = ph ? 1.0f : ry;

  // Layer 1: 8 N-tiles of 16 cols each
  for (int t = 0; t < 8; ++t) {
    const int c = t * 16 + mn;
    v2f b;
    b.x = ph ? W1[2 * LATENT + c] : W1[0 * LATENT + c];
    b.y = ph ? b1[c]              : W1[1 * LATENT + c];
    v8f h8 = {};
    h8 = __builtin_amdgcn_wmma_f32_16x16x4_f32(false, a1, false, b,
                                               (short)0, h8, false, false);
    #pragma unroll
    for (int r = 0; r < 8; ++r) {
      myhid[(r + 8 * ph) * 136 + c] = fmaxf(h8[r], 0.0f);
    }
  }

  // Layer 2: A from LDS (row = edge, K = 128), B = W2 zero-padded to 16 cols
  const int cc = mn;                   // padded output col; valid when < HEADS
  const int c4 = cc & 3;               // safe address for padded lanes
  const float cmask = (cc < HEADS) ? 1.0f : 0.0f;
  v8f acc = {};
  for (int k0 = 0; k0 < LATENT; k0 += 4) {
    v2f a = *(const v2f*)(myhid + mn * 136 + k0 + r0);
    v2f b;
    b.x = W2[(k0 + r0) * HEADS + c4] * cmask;
    b.y = W2[(k0 + r0 + 1) * HEADS + c4] * cmask;
    acc = __builtin_amdgcn_wmma_f32_16x16x4_f32(false, a, false, b,
                                                (short)0, acc, false, false);
  }
  if (cc < HEADS) {
    const float bb = b2[cc];
    #pragma unroll
    for (int r = 0; r < 8; ++r) {
      const int row = e0 + r + 8 * ph;
      logits[row * HEADS + cc] = acc[r] + bb;
    }
  }
  // histogram of edges per destination query (one lane per edge)
  if (lane < 16) atomicAdd(&cnt[d], 1);
}

// ---------------------------------------------------------------------------
// Kernel 0: zero the histogram (d_ws is poisoned, not re-zeroed by harness)
// ---------------------------------------------------------------------------
__global__ void zero_cnt_kernel(int* __restrict__ cnt, int n)
{
  int i = blockIdx.x * blockDim.x + threadIdx.x;
  if (i < n) cnt[i] = 0;
}

// ---------------------------------------------------------------------------
// Kernel 3: single-workgroup exclusive scan cnt -> rowptr; also zeroes cursor.
// ---------------------------------------------------------------------------
__global__ __launch_bounds__(1024) void scan_kernel(
    const int* __restrict__ cnt, int* __restrict__ rowptr,
    int* __restrict__ cursor, int nq)
{
  __shared__ int buf[1024];
  __shared__ int carry;
  const int tid = threadIdx.x;
  if (tid == 0) carry = 0;
  for (int i = tid; i < nq; i += 1024) cursor[i] = 0;
  __syncthreads();
  for (int base = 0; base < nq; base += 1024) {
    const int x = (base + tid < nq) ? cnt[base + tid] : 0;
    buf[tid] = x;
    __syncthreads();
    for (int off = 1; off < 1024; off <<= 1) {
      const int val = (tid >= off) ? buf[tid - off] : 0;
      __syncthreads();
      buf[tid] += val;
      __syncthreads();
    }
    if (base + tid < nq) rowptr[base + tid] = carry + buf[tid] - x;
    __syncthreads();
    if (tid == 0) carry += buf[1023];
    __syncthreads();
  }
  if (tid == 0) rowptr[nq] = carry;
}

// ---------------------------------------------------------------------------
// Kernel 4: scatter edge ids into CSR buckets.
// ---------------------------------------------------------------------------
__global__ void scatter_kernel(const int* __restrict__ dst,
                               const int* __restrict__ rowptr,
                               int* __restrict__ cursor,
                               int* __restrict__ eid)
{
  int e = blockIdx.x * blockDim.x + threadIdx.x;
  if (e < N_EDGE) {
    const int d = dst[e];
    const int slot = atomicAdd(&cursor[d], 1);
    eid[rowptr[d] + slot] = e;
  }
}

// ---------------------------------------------------------------------------
// Kernel 5: one wave per query. Per-head stable softmax over its edges,
// weighted coalesced float4 gather of v[src], no output atomics.
// Lane l covers cols [4l, 4l+4) -> head l>>3.
// ---------------------------------------------------------------------------
__global__ __launch_bounds__(256) void query_gather_kernel(
    const int* __restrict__ rowptr, const int* __restrict__ eid,
    const int* __restrict__ src, const float* __restrict__ logits,
    const float* __restrict__ v, float* __restrict__ out)
{
  const int wave = threadIdx.x >> 5;
  const int lane = threadIdx.x & 31;
  const int q = blockIdx.x * 8 + wave;
  const int h = lane >> 3;
  const int beg = rowptr[q];
  const int end = rowptr[q + 1];

  float mx = -3.0e38f;
  for (int i = beg; i < end; ++i) {
    const int e = eid[i];
    mx = fmaxf(mx, logits[e * HEADS + h]);
  }
  v4f acc = {};
  float ssum = 0.0f;
  for (int i = beg; i < end; ++i) {
    const int e = eid[i];
    const float w = __expf(logits[e * HEADS + h] - mx);
    ssum += w;
    const v4f vv = *(const v4f*)(v + (size_t)src[e] * LATENT + lane * 4);
    acc += vv * w;
  }
  const float inv = (ssum > 0.0f) ? (1.0f / ssum) : 0.0f;
  *(v4f*)(out + (size_t)q * LATENT + lane * 4) = acc * inv;
}

// ---------------------------------------------------------------------------
// Launch
// ---------------------------------------------------------------------------
extern "C" void kernel_launch(void* const* d_in, const int* in_sizes, int n_in,
                              void* d_out, int out_size, void* d_ws, size_t ws_size,
                              hipStream_t stream)
{
  const float* h_obs     = (const float*)d_in[0];
  const float* pos_obs   = (const float*)d_in[1];
  const float* pos_query = (const float*)d_in[2];
  const int*   src       = (const int*)d_in[3];
  const int*   dst       = (const int*)d_in[4];
  const float* Wv        = (const float*)d_in[5];
  const float* bv        = (const float*)d_in[6];
  const float* W1        = (const float*)d_in[7];
  const float* b1        = (const float*)d_in[8];
  const float* W2        = (const float*)d_in[9];
  const float* b2        = (const float*)d_in[10];

  char* ws = (char*)d_ws;
  float* v      = (float*)(ws + 0);          // N_OBS*128 f32 = 51,200,000 B
  float* logits = (float*)(ws + 51200000);   // E*4 f32      = 12,800,000 B
  int*   eid    = (int*)  (ws + 64000000);   // E int        =  3,200,000 B
  int*   cnt    = (int*)  (ws + 67200000);   // N_QRY int
  int*   rowptr = (int*)  (ws + 67400192);   // N_QRY+1 int
  int*   cursor = (int*)  (ws + 67600384);   // N_QRY int

  zero_cnt_kernel<<<dim3((N_QRY + 255) / 256), dim3(256), 0, stream>>>(cnt, N_QRY);
  value_proj_kernel<<<dim3(N_OBS / 16), dim3(256), 0, stream>>>(h_obs, Wv, bv, v);
  edge_mlp_kernel<<<dim3(N_EDGE / 128), dim3(256), 0, stream>>>(
      pos_obs, pos_query, src, dst, W1, b1, W2, b2, logits, cnt);
  scan_kernel<<<dim3(1), dim3(1024), 0, stream>>>(cnt, rowptr, cursor, N_QRY);
  scatter_kernel<<<dim3((N_EDGE + 255) / 256), dim3(256), 0, stream>>>(
      dst, rowptr, cursor, eid);
  query_gather_kernel<<<dim3(N_QRY / 8), dim3(256), 0, stream>>>(
      rowptr, eid, src, logits, v, (float*)d_out);
}